// TemporalGNN_53953379173119
// MI455X (gfx1250) — compile-verified
//
#include <hip/hip_runtime.h>

// TGCN (TemporalGNN) for MI455X / gfx1250.
// Strategy: exploit linearity of GCNConv -> do the (E+N)-edge scatter ONCE on
// raw features Xagg[N,96] = normAdj @ x, then all convs/gates become dense
// per-node-tile GEMMs done with fp32 WMMA (V_WMMA_F32_16X16X4_F32, wave32).
// The block's Xagg tile (64 nodes x 96 f32 = 24KB, contiguous) is DMA'd into
// LDS with the Tensor Data Mover (tensor_load_to_lds + s_wait_tensorcnt).

#define F_IN 8
#define T_STEPS 12
#define CH 32
#define FT (F_IN * T_STEPS) /* 96 */

typedef __attribute__((ext_vector_type(2))) float v2f;
typedef __attribute__((ext_vector_type(8))) float v8f;
typedef __attribute__((ext_vector_type(4))) unsigned int v4u;
typedef __attribute__((ext_vector_type(4))) int v4i;
typedef __attribute__((ext_vector_type(8))) int v8i;

__device__ __forceinline__ v8f wmma4(v2f a, v2f b, v8f c) {
  // D = A(16x4,f32) * B(4x16,f32) + C(16x16,f32)
  return __builtin_amdgcn_wmma_f32_16x16x4_f32(false, a, false, b, (short)0, c,
                                               false, false);
}

__device__ __forceinline__ v8f sigmoid8(v8f x) {
  v8f y;
#pragma unroll
  for (int i = 0; i < 8; ++i) y[i] = 1.0f / (1.0f + __expf(-x[i]));
  return y;
}

// tanh(x) = 2*sigmoid(2x) - 1  (exact identity; fast __expf form)
__device__ __forceinline__ v8f tanh8(v8f x) {
  v8f y;
#pragma unroll
  for (int i = 0; i < 8; ++i)
    y[i] = 2.0f / (1.0f + __expf(-2.0f * x[i])) - 1.0f;
  return y;
}

// ---------------- graph preprocessing ----------------

__global__ void k_init_deg(float* deg, int n) {
  int i = blockIdx.x * blockDim.x + threadIdx.x;
  if (i < n) deg[i] = 1.0f;  // self-loop weight
}

__global__ void k_scatter_deg(const int* ei, const float* ew, float* deg,
                              int ne) {
  int i = blockIdx.x * blockDim.x + threadIdx.x;
  if (i < ne) atomicAdd(&deg[ei[ne + i]], ew[i]);  // segment over col
}

__global__ void k_dinv(float* deg, int n) {
  int i = blockIdx.x * blockDim.x + threadIdx.x;
  if (i < n) deg[i] = rsqrtf(deg[i]);  // deg >= 1 always
}

// Xagg init with the self-loop contribution: dinv[i]^2 * x[i]
__global__ void k_self(const float* __restrict__ x,
                       const float* __restrict__ dinv, float* xagg,
                       int total) {
  int i = blockIdx.x * blockDim.x + threadIdx.x;
  if (i >= total) return;
  int node = i / FT;
  float d = dinv[node];
  xagg[i] = d * d * x[i];
}

// one scatter pass over all 96 channels (8 features x 12 timesteps)
__global__ void k_edges(const int* __restrict__ ei,
                        const float* __restrict__ ew,
                        const float* __restrict__ dinv,
                        const float* __restrict__ x, float* xagg, int ne) {
  int t = blockIdx.x * blockDim.x + threadIdx.x;
  int e = t >> 2;
  int part = t & 3;
  if (e >= ne) return;
  int r = ei[e];
  int c = ei[ne + e];
  float w = dinv[r] * ew[e] * dinv[c];
  const float* xs = x + (long)r * FT + part * 24;
  float* xd = xagg + (long)c * FT + part * 24;
#pragma unroll
  for (int k = 0; k < 24; ++k) atomicAdd(&xd[k], w * xs[k]);
}

// ---------------- WMMA helpers ----------------
// A (16x4 f32): lane m=l%16, h=l/16 holds K = {k0+2h, k0+2h+1} in {a.x,a.y}
// B (4x16 f32): lane n=l%16, h=l/16 holds K rows {k0+2h, k0+2h+1} at col n
// C/D (16x16):  vgpr r, lane l -> M = r + 8*(l/16), N = l%16

__device__ __forceinline__ v2f ldB(const float* W, int ldw, int k0, int n,
                                   int h) {
  v2f b;
  b.x = W[(k0 + 2 * h) * ldw + n];
  b.y = W[(k0 + 2 * h + 1) * ldw + n];
  return b;
}

__device__ __forceinline__ void store_tile(float* buf, int c0, v8f d, int m,
                                           int h) {
#pragma unroll
  for (int r = 0; r < 8; ++r) buf[(r + 8 * h) * 64 + c0 + m] = d[r];
}

// [16x64] (LDS staging, row-major, stride 64) @ L[64x32] + bias
__device__ __forceinline__ v8f gate_gemm(const float* buf, const float* L,
                                         float bias, int n0, int m, int h) {
  v8f c;
#pragma unroll
  for (int r = 0; r < 8; ++r) c[r] = bias;
#pragma unroll
  for (int k0 = 0; k0 < 64; k0 += 4) {
    v2f a;
    a.x = buf[m * 64 + k0 + 2 * h];
    a.y = buf[m * 64 + k0 + 2 * h + 1];
    c = wmma4(a, ldB(L, CH, k0, n0 + m, h), c);
  }
  return c;
}

// conv tile: A (Xagg slice, preloaded) @ W[8x32] + bias
__device__ __forceinline__ v8f conv_gemm(v2f a0, v2f a1, const float* W,
                                         float bias, int n0, int m, int h) {
  v8f c;
#pragma unroll
  for (int r = 0; r < 8; ++r) c[r] = bias;
  c = wmma4(a0, ldB(W, CH, 0, n0 + m, h), c);
  c = wmma4(a1, ldB(W, CH, 4, n0 + m, h), c);
  return c;
}

// ---------------- fused GRU over time ----------------

__global__ __launch_bounds__(128) void k_tgcn(
    const float* __restrict__ xagg, const float* __restrict__ Wz,
    const float* __restrict__ bz, const float* __restrict__ Wr,
    const float* __restrict__ br, const float* __restrict__ Wh,
    const float* __restrict__ bh, const float* __restrict__ Lz,
    const float* __restrict__ Lzb, const float* __restrict__ Lr,
    const float* __restrict__ Lrb, const float* __restrict__ Lh,
    const float* __restrict__ Lhb, const float* __restrict__ att,
    const float* __restrict__ linw, const float* __restrict__ linb,
    float* __restrict__ out, int n) {
  __shared__ float sW[3][F_IN][CH];     // 768
  __shared__ float sWb[3][CH];          // 96
  __shared__ float sL[3][2 * CH][CH];   // 6144
  __shared__ float sLb[3][CH];          // 96
  __shared__ float sLin[CH][16];        // 512 (cols 12..15 zero-padded)
  __shared__ float sLinb[16];
  __shared__ float sAtt[16];
  __shared__ float sStage[4][16][64];   // per-wave staging, 16 KB
  __shared__ float sXagg[64 * FT];      // block's Xagg tile, 24 KB (TDM dest)

  const int tid = threadIdx.x;
  const int wv = tid >> 5;
  const int lane = tid & 31;
  const int nb0 = blockIdx.x * 64;  // block's node base (always < n)

  // ---- TDM: DMA this block's Xagg tile (contiguous) into LDS ----
  if (wv == 0) {
    unsigned long long ga =
        (unsigned long long)(const void*)(xagg + (long)nb0 * FT);
    unsigned ldsOff = (unsigned)(unsigned long long)(void*)&sXagg[0];
    unsigned td0 = (unsigned)(n - nb0) * FT;  // valid elements (OOB reads -> 0)
    v4u g0;
    g0.x = 1u;                                   // count=1, user descriptor
    g0.y = ldsOff;                               // lds_addr [63:32]
    g0.z = (unsigned)ga;                         // global_addr [95:64]
    g0.w = (unsigned)(ga >> 32) | 0x80000000u;   // ga[56:32] + type=2 [127:126]
    v8i g1;
    g1[0] = (int)(2u << 16);                     // data_size=4B, no multicast
    g1[1] = (int)((td0 & 0xFFFFu) << 16);        // tensor_dim0 lo16 @ [63:48]
    g1[2] = (int)((td0 >> 16) | (1u << 16));     // tensor_dim0 hi16, dim1=1
    g1[3] = (int)((unsigned)(64 * FT) << 16);    // tile_dim0 = 6144 @ [127:112]
    g1[4] = 0;                                   // tile_dim1=0, tile_dim2=0
    g1[5] = (int)td0;                            // tensor_dim0_stride lo32
    g1[6] = 0;                                   // stride hi, dim1_stride lo
    g1[7] = 0;
    v4i gz = {0, 0, 0, 0};
#if __clang_major__ >= 23
    v8i gz8 = {0, 0, 0, 0, 0, 0, 0, 0};
    __builtin_amdgcn_tensor_load_to_lds(g0, g1, gz, gz, gz8, 0);
#else
    __builtin_amdgcn_tensor_load_to_lds(g0, g1, gz, gz, 0);
#endif
  }

  // ---- cooperative weight staging (overlaps with the TDM transfer) ----
  for (int i = tid; i < 3 * F_IN * CH; i += blockDim.x) {
    int g = i / (F_IN * CH), j = i % (F_IN * CH);
    (&sW[0][0][0])[i] = (g == 0 ? Wz : g == 1 ? Wr : Wh)[j];
  }
  for (int i = tid; i < 3 * CH; i += blockDim.x) {
    int g = i / CH, j = i % CH;
    (&sWb[0][0])[i] = (g == 0 ? bz : g == 1 ? br : bh)[j];
    (&sLb[0][0])[i] = (g == 0 ? Lzb : g == 1 ? Lrb : Lhb)[j];
  }
  for (int i = tid; i < 3 * 2 * CH * CH; i += blockDim.x) {
    int g = i / (2 * CH * CH), j = i % (2 * CH * CH);
    (&sL[0][0][0])[i] = (g == 0 ? Lz : g == 1 ? Lr : Lh)[j];
  }
  for (int i = tid; i < CH * 16; i += blockDim.x) {
    int r = i / 16, c = i % 16;
    sLin[r][c] = (c < T_STEPS) ? linw[r * T_STEPS + c] : 0.0f;
  }
  if (tid < 16) sLinb[tid] = (tid < T_STEPS) ? linb[tid] : 0.0f;
  if (tid < 16) sAtt[tid] = (tid < T_STEPS) ? att[tid] : 0.0f;

  if (wv == 0) __builtin_amdgcn_s_wait_tensorcnt(0);  // TENSORcnt is per-wave
  __syncthreads();

  const int m = lane & 15;  // row (A/C) or col (B/C)
  const int h = lane >> 4;  // lane half
  const int tile = blockIdx.x * 4 + wv;
  const int nb = tile * 16;
  if (nb >= n) return;

  float* buf = &sStage[wv][0][0];

  // softmax(att) normalizers (T=12, redundant per thread, cheap)
  float amax = sAtt[0];
  for (int t = 1; t < T_STEPS; ++t) amax = fmaxf(amax, sAtt[t]);
  float ssum = 0.0f;
  for (int t = 0; t < T_STEPS; ++t) ssum += __expf(sAtt[t] - amax);
  const float invsum = 1.0f / ssum;

  // per-lane A-operand row now lives in LDS (TDM-filled tile)
  const float* xrow = &sXagg[(wv * 16 + m) * FT];

  v8f H0, H1, Ha0, Ha1;
#pragma unroll
  for (int r = 0; r < 8; ++r) { H0[r] = H1[r] = Ha0[r] = Ha1[r] = 0.0f; }

  for (int t = 0; t < T_STEPS; ++t) {
    // A tiles for all three convs (shared!): features [0..3] and [4..7]
    v2f a0, a1;
    a0.x = xrow[(0 + 2 * h) * T_STEPS + t];
    a0.y = xrow[(1 + 2 * h) * T_STEPS + t];
    a1.x = xrow[(4 + 2 * h) * T_STEPS + t];
    a1.y = xrow[(5 + 2 * h) * T_STEPS + t];

    // ---- Z gate ----
    {
      v8f c0 = conv_gemm(a0, a1, &sW[0][0][0], sWb[0][m], 0, m, h);
      v8f c1 = conv_gemm(a0, a1, &sW[0][0][0], sWb[0][16 + m], 16, m, h);
      store_tile(buf, 0, c0, m, h);
      store_tile(buf, 16, c1, m, h);
    }
    store_tile(buf, 32, H0, m, h);
    store_tile(buf, 48, H1, m, h);
    v8f z0 = sigmoid8(gate_gemm(buf, &sL[0][0][0], sLb[0][m], 0, m, h));
    v8f z1 = sigmoid8(gate_gemm(buf, &sL[0][0][0], sLb[0][16 + m], 16, m, h));

    // ---- R gate (H already staged in cols 32..63) ----
    {
      v8f c0 = conv_gemm(a0, a1, &sW[1][0][0], sWb[1][m], 0, m, h);
      v8f c1 = conv_gemm(a0, a1, &sW[1][0][0], sWb[1][16 + m], 16, m, h);
      store_tile(buf, 0, c0, m, h);
      store_tile(buf, 16, c1, m, h);
    }
    v8f r0 = sigmoid8(gate_gemm(buf, &sL[1][0][0], sLb[1][m], 0, m, h));
    v8f r1 = sigmoid8(gate_gemm(buf, &sL[1][0][0], sLb[1][16 + m], 16, m, h));

    // ---- candidate H~ : [conv_h | H*R] ----
    {
      v8f c0 = conv_gemm(a0, a1, &sW[2][0][0], sWb[2][m], 0, m, h);
      v8f c1 = conv_gemm(a0, a1, &sW[2][0][0], sWb[2][16 + m], 16, m, h);
      store_tile(buf, 0, c0, m, h);
      store_tile(buf, 16, c1, m, h);
    }
    {
      v8f hr0, hr1;
#pragma unroll
      for (int r = 0; r < 8; ++r) {
        hr0[r] = H0[r] * r0[r];
        hr1[r] = H1[r] * r1[r];
      }
      store_tile(buf, 32, hr0, m, h);
      store_tile(buf, 48, hr1, m, h);
    }
    v8f t0 = tanh8(gate_gemm(buf, &sL[2][0][0], sLb[2][m], 0, m, h));
    v8f t1 = tanh8(gate_gemm(buf, &sL[2][0][0], sLb[2][16 + m], 16, m, h));

    // ---- GRU update + attention accumulate ----
    const float p = __expf(sAtt[t] - amax) * invsum;
#pragma unroll
    for (int r = 0; r < 8; ++r) {
      H0[r] = z0[r] * H0[r] + (1.0f - z0[r]) * t0[r];
      H1[r] = z1[r] * H1[r] + (1.0f - z1[r]) * t1[r];
      Ha0[r] += p * H0[r];
      Ha1[r] += p * H1[r];
    }
  }

  // ---- final: relu(Hacc) @ lin_w[32x12] + lin_b ----
  {
    v8f q0, q1;
#pragma unroll
    for (int r = 0; r < 8; ++r) {
      q0[r] = fmaxf(Ha0[r], 0.0f);
      q1[r] = fmaxf(Ha1[r], 0.0f);
    }
    store_tile(buf, 0, q0, m, h);
    store_tile(buf, 16, q1, m, h);
  }
  v8f o;
#pragma unroll
  for (int r = 0; r < 8; ++r) o[r] = sLinb[m];
#pragma unroll
  for (int k0 = 0; k0 < 2 * 16; k0 += 4) {
    v2f a;
    a.x = buf[m * 64 + k0 + 2 * h];
    a.y = buf[m * 64 + k0 + 2 * h + 1];
    o = wmma4(a, ldB(&sLin[0][0], 16, k0, m, h), o);
  }
#pragma unroll
  for (int r = 0; r < 8; ++r) {
    int node = nb + r + 8 * h;
    if (node < n && m < T_STEPS) out[node * T_STEPS + m] = o[r];
  }
}

// ---------------- launch ----------------

extern "C" void kernel_launch(void* const* d_in, const int* in_sizes, int n_in,
                              void* d_out, int out_size, void* d_ws,
                              size_t ws_size, hipStream_t stream) {
  const float* x = (const float*)d_in[0];
  const int* ei = (const int*)d_in[1];
  const float* ew = (const float*)d_in[2];
  const float* Wz = (const float*)d_in[3];
  const float* bz = (const float*)d_in[4];
  const float* Wr = (const float*)d_in[5];
  const float* br = (const float*)d_in[6];
  const float* Wh = (const float*)d_in[7];
  const float* bh = (const float*)d_in[8];
  const float* Lz = (const float*)d_in[9];
  const float* Lzb = (const float*)d_in[10];
  const float* Lr = (const float*)d_in[11];
  const float* Lrb = (const float*)d_in[12];
  const float* Lh = (const float*)d_in[13];
  const float* Lhb = (const float*)d_in[14];
  const float* att = (const float*)d_in[15];
  const float* linw = (const float*)d_in[16];
  const float* linb = (const float*)d_in[17];
  float* out = (float*)d_out;

  const int n = in_sizes[0] / FT;  // 50000
  const int e = in_sizes[2];       // 800000

  // workspace: deg/dinv [n] floats, then Xagg [n*96] floats (~19.4 MB total)
  float* d_deg = (float*)d_ws;
  size_t off = (((size_t)n * sizeof(float)) + 255) & ~(size_t)255;
  float* d_xagg = (float*)((char*)d_ws + off);

  k_init_deg<<<(n + 255) / 256, 256, 0, stream>>>(d_deg, n);
  k_scatter_deg<<<(e + 255) / 256, 256, 0, stream>>>(ei, ew, d_deg, e);
  k_dinv<<<(n + 255) / 256, 256, 0, stream>>>(d_deg, n);
  k_self<<<(n * FT + 255) / 256, 256, 0, stream>>>(x, d_deg, d_xagg, n * FT);
  k_edges<<<(e * 4 + 255) / 256, 256, 0, stream>>>(ei, ew, d_deg, x, d_xagg, e);

  const int tiles = (n + 15) / 16;     // 3125 16-node tiles, 1 per wave
  const int blocks = (tiles + 3) / 4;  // 4 waves per block
  k_tgcn<<<blocks, 128, 0, stream>>>(d_xagg, Wz, bz, Wr, br, Wh, bh, Lz, Lzb,
                                     Lr, Lrb, Lh, Lhb, att, linw, linb, out, n);
}